// GCN_18528488915141
// MI455X (gfx1250) — compile-verified
//
#include <hip/hip_runtime.h>

typedef __attribute__((ext_vector_type(2))) float v2f;
typedef __attribute__((ext_vector_type(8))) float v8f;

#define N_NODES 50000
#define N_EDGES 800000
#define TILES_M (N_NODES / 16)   // 3125, exact

// ---------------- utility kernels ----------------

__global__ void zero_f32(float* __restrict__ p, int n) {
    int i = blockIdx.x * blockDim.x + threadIdx.x;
    if (i < n) p[i] = 0.0f;
}

__global__ void degree_kernel(const int* __restrict__ src, const int* __restrict__ dst,
                              float* __restrict__ outdeg, float* __restrict__ indeg, int E) {
    int e = blockIdx.x * blockDim.x + threadIdx.x;
    if (e >= E) return;
    atomicAdd(&outdeg[src[e]], 1.0f);
    atomicAdd(&indeg[dst[e]], 1.0f);
}

// in-place: deg -> clamp(deg,1)^-0.5   (applied to both arrays at once, contiguous)
__global__ void norm_kernel(float* __restrict__ d, int n) {
    int i = blockIdx.x * blockDim.x + threadIdx.x;
    if (i >= n) return;
    float v = d[i];
    v = v < 1.0f ? 1.0f : v;
    d[i] = 1.0f / sqrtf(v);
}

// ---------------- dense transform: OUT[N,DO] = H[N,DI] @ W[DI,DO] ----------------
// One wave per 16-row tile; accumulates DO/16 16x16 f32 tiles with
// V_WMMA_F32_16X16X4_F32 (A 16x4 f32, B 4x16 f32, C 16x16 f32).
//
// A layout (ISA 7.12.2, 32-bit A 16x4): lane l holds row M = l%16;
//   lanes 0-15:  VGPR0=K0, VGPR1=K1 ; lanes 16-31: VGPR0=K2, VGPR1=K3
// B layout (4x16, row striped across lanes within a VGPR, mirrored):
//   lanes 0-15:  N=0..15, VGPR0=K0, VGPR1=K1 ; lanes 16-31: VGPR0=K2, VGPR1=K3
// C/D layout: VGPR r -> M=r (lanes 0-15) / M=r+8 (lanes 16-31), N = lane%16.
template <int DI, int DO>
__global__ void gemm_wmma_kernel(const float* __restrict__ H,
                                 const float* __restrict__ W,
                                 float* __restrict__ OUT, int m_tiles) {
    constexpr int NT = DO / 16;
    int wave = (int)((blockIdx.x * blockDim.x + threadIdx.x) >> 5);
    if (wave >= m_tiles) return;
    int lane = threadIdx.x & 31;
    int lan16 = lane & 15;
    int hi = lane >> 4;          // 0 or 1
    int m0 = wave * 16;
    int row = m0 + lan16;

    v8f acc[NT] = {};
    const float* hrow = H + (long)row * DI;

    for (int k0 = 0; k0 < DI; k0 += 4) {
        // A fragment: 8-byte aligned float2 load (k0 mult of 4, 2*hi even)
        v2f a = *(const v2f*)(hrow + k0 + 2 * hi);
#pragma unroll
        for (int t = 0; t < NT; ++t) {
            int col = t * 16 + lan16;
            v2f b;
            b.x = W[(k0 + 2 * hi) * DO + col];
            b.y = W[(k0 + 2 * hi + 1) * DO + col];
            acc[t] = __builtin_amdgcn_wmma_f32_16x16x4_f32(
                false, a, false, b, (short)0, acc[t], false, false);
        }
    }

#pragma unroll
    for (int t = 0; t < NT; ++t) {
#pragma unroll
        for (int r = 0; r < 8; ++r) {
            OUT[(long)(m0 + r + 8 * hi) * DO + t * 16 + lan16] = acc[t][r];
        }
    }
}

// ---------------- message scatter: agg[dst] += tmp[src] * norm_src[src] ----------------
// thread = (edge, channel); consecutive lanes cover consecutive channels of one
// edge -> coalesced gather and coalesced atomic adds (L2-resident agg buffer).
template <int DO>
__global__ void scatter_kernel(const float* __restrict__ tmp, const int* __restrict__ src,
                               const int* __restrict__ dst, const float* __restrict__ nsrc,
                               float* __restrict__ agg) {
    long gid = (long)blockIdx.x * blockDim.x + threadIdx.x;
    if (gid >= (long)N_EDGES * DO) return;
    int e = (int)(gid >> __builtin_ctz(DO));
    int c = (int)(gid & (DO - 1));
    int s = src[e];
    int d = dst[e];
    float v = tmp[s * DO + c] * nsrc[s];
    atomicAdd(&agg[d * DO + c], v);
}

// ---------------- finalize: out = relu(agg * norm_dst + b) ----------------
template <int DO>
__global__ void finalize_kernel(const float* __restrict__ agg, const float* __restrict__ ndst,
                                const float* __restrict__ bias, float* __restrict__ out) {
    int gid = blockIdx.x * blockDim.x + threadIdx.x;
    if (gid >= N_NODES * DO) return;
    int i = gid >> __builtin_ctz(DO);
    int c = gid & (DO - 1);
    float v = agg[gid] * ndst[i] + bias[c];
    out[gid] = v > 0.0f ? v : 0.0f;
}

// ---------------- per-layer driver ----------------
template <int DI, int DO>
static void run_layer(const float* h_in, const float* W, const float* b,
                      float* tmp, float* agg, float* h_out,
                      const float* nsrc, const float* ndst,
                      const int* src, const int* dst, hipStream_t stream) {
    // dense transform
    {
        int threads = 128;                       // 4 waves per block
        int blocks = (TILES_M * 32 + threads - 1) / threads;
        gemm_wmma_kernel<DI, DO><<<blocks, threads, 0, stream>>>(h_in, W, tmp, TILES_M);
    }
    // zero aggregation buffer
    {
        int n = N_NODES * DO;
        zero_f32<<<(n + 255) / 256, 256, 0, stream>>>(agg, n);
    }
    // scatter-add messages
    {
        long total = (long)N_EDGES * DO;
        long blocks = (total + 255) / 256;
        scatter_kernel<DO><<<(int)blocks, 256, 0, stream>>>(tmp, src, dst, nsrc, agg);
    }
    // normalize + bias + relu
    {
        int n = N_NODES * DO;
        finalize_kernel<DO><<<(n + 255) / 256, 256, 0, stream>>>(agg, ndst, b, h_out);
    }
}

extern "C" void kernel_launch(void* const* d_in, const int* in_sizes, int n_in,
                              void* d_out, int out_size, void* d_ws, size_t ws_size,
                              hipStream_t stream) {
    const float* x     = (const float*)d_in[0];
    const int*   edges = (const int*)d_in[1];
    const int*   src   = edges;            // edges[0], row-major (2,E)
    const int*   dst   = edges + N_EDGES;  // edges[1]
    const float* W1 = (const float*)d_in[2];  const float* b1 = (const float*)d_in[3];
    const float* W2 = (const float*)d_in[4];  const float* b2 = (const float*)d_in[5];
    const float* W3 = (const float*)d_in[6];  const float* b3 = (const float*)d_in[7];
    const float* W4 = (const float*)d_in[8];  const float* b4 = (const float*)d_in[9];
    const float* W5 = (const float*)d_in[10]; const float* b5 = (const float*)d_in[11];
    float* out = (float*)d_out;

    // workspace layout (floats):
    //   [0, N)        norm_src (deg computed in place)
    //   [N, 2N)       norm_dst
    //   [2N, 2N+N*128)        bufA
    //   [2N+N*128, 2N+2N*128) bufB
    float* ws   = (float*)d_ws;
    float* nsrc = ws;
    float* ndst = ws + N_NODES;
    float* bufA = ws + 2 * N_NODES;
    float* bufB = bufA + (long)N_NODES * 128;

    // degrees -> norms
    zero_f32<<<(2 * N_NODES + 255) / 256, 256, 0, stream>>>(nsrc, 2 * N_NODES);
    degree_kernel<<<(N_EDGES + 255) / 256, 256, 0, stream>>>(src, dst, nsrc, ndst, N_EDGES);
    norm_kernel<<<(2 * N_NODES + 255) / 256, 256, 0, stream>>>(nsrc, 2 * N_NODES);

    // 5 GCN layers, ping-ponging bufA/bufB (h_out overwrites the tmp buffer;
    // stream ordering guarantees agg (= previous h_in) is free after the GEMM).
    run_layer<128, 128>(x,    W1, b1, bufA, bufB, bufA, nsrc, ndst, src, dst, stream); // h1 in A
    run_layer<128,  64>(bufA, W2, b2, bufB, bufA, bufB, nsrc, ndst, src, dst, stream); // h2 in B
    run_layer< 64,  32>(bufB, W3, b3, bufA, bufB, bufA, nsrc, ndst, src, dst, stream); // h3 in A
    run_layer< 32,  16>(bufA, W4, b4, bufB, bufA, bufB, nsrc, ndst, src, dst, stream); // h4 in B
    run_layer< 16,  16>(bufB, W5, b5, bufA, bufB, out,  nsrc, ndst, src, dst, stream); // -> d_out
    (void)in_sizes; (void)n_in; (void)out_size; (void)ws_size;
}